// RecurrentCA_50998441672731
// MI455X (gfx1250) — compile-verified
//
#include <hip/hip_runtime.h>
#include <math.h>

// ---------------- problem constants ----------------
#define C_CH   64
#define T_LEN  1024
#define M_Q    16
#define M_M    16
#define D_K    64
#define MOMv   0.05f
#define EPS_V  1e-12f

// ---------------- mapping ----------------
#define NBLK   16        // workgroups participating in the per-step barrier
#define WPB    4         // waves (== channels) per workgroup
#define NTHR   (WPB*32)  // wave32

// ---------------- per-wave LDS layout (dwords), padded pitches ----------------
#define ATT_PITCH 18     // 16x16 att, conflict-free A-gather
#define KQ_PITCH  68     // 16x64 matrices, 4n+kk bank pattern
#define V_PITCH   72     // v B-gather keeps lane halves on disjoint banks
#define ATT_SZ (16*ATT_PITCH)          // 288
#define PB_SZ  (16*KQ_PITCH)           // 1088  (p = q - out)
#define KB_SZ  (2*16*KQ_PITCH)         // 2176  (double buffered)
#define QB_SZ  (2*16*KQ_PITCH)         // 2176
#define VB_SZ  (2*16*V_PITCH)          // 2304
#define WAVE_LDS (ATT_SZ+PB_SZ+KB_SZ+QB_SZ+VB_SZ)   // 8032 dwords / wave
#define BLK_LDS  (WPB*WAVE_LDS + WPB*16 + 16)        // + wgSum[4][16] + nrm[16]

typedef float v2f __attribute__((ext_vector_type(2)));
typedef float v8f __attribute__((ext_vector_type(8)));
typedef int   vi4 __attribute__((vector_size(16)));   // matches builtin param type

#if __has_builtin(__builtin_amdgcn_global_load_async_to_lds_b128)
#define USE_ASYNC 1
#else
#define USE_ASYNC 0
#endif

#if USE_ASYNC
__device__ __forceinline__ void async_cp_b128(const float* g, float* l) {
    // ASYNCcnt-tracked DMA: 16B per lane, per-lane LDS destination address.
    __builtin_amdgcn_global_load_async_to_lds_b128(
        (vi4*)g, (vi4*)l, 0 /*imm offset*/, 0 /*cpol*/);
}
#endif

__global__ void __launch_bounds__(NTHR)
rca_kernel(const float* __restrict__ Kg, const float* __restrict__ Qg,
           const float* __restrict__ Vg, float* __restrict__ Og,
           float* __restrict__ partial,   // [2][NBLK][16] floats
           unsigned* __restrict__ cnt)    // monotonic arrive counter
{
    extern __shared__ float smem[];
    const int tid  = threadIdx.x;
    const int w    = tid >> 5;     // wave id == local channel
    const int lane = tid & 31;
    const int hi   = lane >> 4;    // lane half (WMMA layout)
    const int ln   = lane & 15;
    const int c    = blockIdx.x * WPB + w;

    float* ATT = smem + w * WAVE_LDS;
    float* PB  = ATT + ATT_SZ;
    float* KB  = PB  + PB_SZ;
    float* QB  = KB  + KB_SZ;
    float* VB  = QB  + QB_SZ;
    float* wgSum = smem + WPB * WAVE_LDS;  // [WPB][16]
    float* nrmb  = wgSum + WPB * 16;       // [16]

    const float* Kc = Kg + (size_t)c * T_LEN * M_M * D_K;
    const float* Qc = Qg + (size_t)c * T_LEN * M_Q * D_K;
    const float* Vc = Vg + (size_t)c * T_LEN * M_M * D_K;
    float*       Oc = Og + (size_t)c * T_LEN * M_Q * D_K;

    // att0 = ones (canonical row-major in LDS; lane holds col ln, rows r+8*hi)
#pragma unroll
    for (int r = 0; r < 8; ++r)
        ATT[(r + 8*hi) * ATT_PITCH + ln] = 1.0f;

    float s_[8];
#pragma unroll
    for (int r = 0; r < 8; ++r) s_[r] = 0.0f;
    float pw = 1.0f;               // (1-m)^t running power

    const int colc = 4*ln;

#if USE_ASYNC
    // issue step-0 tile DMAs: 8 x b128 per matrix = 4KB each, padded-pitch dest
    {
        float* kb = KB; float* qb = QB; float* vb = VB;
#pragma unroll
        for (int i = 0; i < 8; ++i) {
            const int row = 2*i + hi;
            const int goff = 4*lane + 128*i;
            async_cp_b128(Kc + goff, kb + row*KQ_PITCH + colc);
            async_cp_b128(Qc + goff, qb + row*KQ_PITCH + colc);
            async_cp_b128(Vc + goff, vb + row*V_PITCH  + colc);
        }
    }
#else
    // fallback: stage step 0 into registers
    float4 pk[8], pq[8], pv[8];
    {
        const float4* gk = (const float4*)Kc;
        const float4* gq = (const float4*)Qc;
        const float4* gv = (const float4*)Vc;
#pragma unroll
        for (int i = 0; i < 8; ++i) {
            pk[i] = gk[lane + 32*i];
            pq[i] = gq[lane + 32*i];
            pv[i] = gv[lane + 32*i];
        }
    }
#endif

    for (int t = 0; t < T_LEN; ++t) {
        const int buf = t & 1;
        float* kb = KB + buf * (16*KQ_PITCH);
        float* qb = QB + buf * (16*KQ_PITCH);
        float* vb = VB + buf * (16*V_PITCH);

#if USE_ASYNC
        // wait for step-t tiles to land in LDS, then kick off step t+1 into the
        // other buffer (its step t-1 readers all completed inside step t-1).
        asm volatile("s_wait_asynccnt 0x0" ::: "memory");
        if (t + 1 < T_LEN) {
            float* kb2 = KB + (buf ^ 1) * (16*KQ_PITCH);
            float* qb2 = QB + (buf ^ 1) * (16*KQ_PITCH);
            float* vb2 = VB + (buf ^ 1) * (16*V_PITCH);
            const float* Kt = Kc + (size_t)(t+1) * (M_M*D_K);
            const float* Qt = Qc + (size_t)(t+1) * (M_Q*D_K);
            const float* Vt = Vc + (size_t)(t+1) * (M_M*D_K);
#pragma unroll
            for (int i = 0; i < 8; ++i) {
                const int row = 2*i + hi;
                const int goff = 4*lane + 128*i;
                async_cp_b128(Kt + goff, kb2 + row*KQ_PITCH + colc);
                async_cp_b128(Qt + goff, qb2 + row*KQ_PITCH + colc);
                async_cp_b128(Vt + goff, vb2 + row*V_PITCH  + colc);
            }
        }
#else
        // commit staged tiles (step t) to padded LDS buffers
#pragma unroll
        for (int i = 0; i < 8; ++i) {
            const int row = 2*i + hi;
            *(float4*)(kb + row*KQ_PITCH + colc) = pk[i];
            *(float4*)(qb + row*KQ_PITCH + colc) = pq[i];
            *(float4*)(vb + row*V_PITCH  + colc) = pv[i];
        }
        // prefetch step t+1 into registers
        if (t + 1 < T_LEN) {
            const float4* gk = (const float4*)(Kc + (size_t)(t+1) * (M_M*D_K));
            const float4* gq = (const float4*)(Qc + (size_t)(t+1) * (M_Q*D_K));
            const float4* gv = (const float4*)(Vc + (size_t)(t+1) * (M_M*D_K));
#pragma unroll
            for (int i = 0; i < 8; ++i) {
                pk[i] = gk[lane + 32*i];
                pq[i] = gq[lane + 32*i];
                pv[i] = gv[lane + 32*i];
            }
        }
#endif

        // ---- out = att(16x16) @ v(16x64): 4 N-tiles, K=16 in chunks of 4 ----
        v8f acc[4];
#pragma unroll
        for (int nt = 0; nt < 4; ++nt) acc[nt] = (v8f){0,0,0,0,0,0,0,0};
#pragma unroll
        for (int kk = 0; kk < 16; kk += 4) {
            v2f a;                                   // A: lane row=ln, K pair kk+2*hi
            a.x = ATT[ln*ATT_PITCH + kk + 2*hi];
            a.y = ATT[ln*ATT_PITCH + kk + 2*hi + 1];
#pragma unroll
            for (int nt = 0; nt < 4; ++nt) {
                v2f b;                               // B: row kk+2*hi of v, col 16*nt+ln
                b.x = vb[(kk + 2*hi    )*V_PITCH + 16*nt + ln];
                b.y = vb[(kk + 2*hi + 1)*V_PITCH + 16*nt + ln];
                acc[nt] = __builtin_amdgcn_wmma_f32_16x16x4_f32(
                    false, a, false, b, (short)0, acc[nt], false, false);
            }
        }

        // ---- p = q - out ; write out_t to global ; stage p in LDS ----
#pragma unroll
        for (int nt = 0; nt < 4; ++nt) {
#pragma unroll
            for (int r = 0; r < 8; ++r) {
                const int row = r + 8*hi, col = 16*nt + ln;
                const float o = acc[nt][r];
                Oc[(size_t)t * (M_Q*D_K) + row*D_K + col] = o;
                PB[row*KQ_PITCH + col] = qb[row*KQ_PITCH + col] - o;
            }
        }

        // ---- raw = p(16x64) @ k^T(64x16): K=64 in chunks of 4 ----
        v8f raw = (v8f){0,0,0,0,0,0,0,0};
#pragma unroll
        for (int kk = 0; kk < 64; kk += 4) {
            v2f a, b;
            a.x = PB[ln*KQ_PITCH + kk + 2*hi];       // A row=ln of p
            a.y = PB[ln*KQ_PITCH + kk + 2*hi + 1];
            b.x = kb[ln*KQ_PITCH + kk + 2*hi];       // B[kk][n] = k[n][kk] -> same gather
            b.y = kb[ln*KQ_PITCH + kk + 2*hi + 1];
            raw = __builtin_amdgcn_wmma_f32_16x16x4_f32(
                false, a, false, b, (short)0, raw, false, false);
        }

        // ---- EMA + bias correction; per-row (q) partial sum of squares ----
        pw *= (1.0f - MOMv);
        const float corr = 1.0f / (1.0f - pw);
        float ah[8];
#pragma unroll
        for (int r = 0; r < 8; ++r) {
            s_[r] = (1.0f - MOMv) * s_[r] + MOMv * raw[r];
            ah[r] = s_[r] * corr;
        }
#pragma unroll
        for (int r = 0; r < 8; ++r) {
            float v2 = ah[r] * ah[r];
            v2 += __shfl_xor(v2, 1, 16);             // reduce over cols within half
            v2 += __shfl_xor(v2, 2, 16);
            v2 += __shfl_xor(v2, 4, 16);
            v2 += __shfl_xor(v2, 8, 16);
            if (ln == 0) wgSum[w*16 + r + 8*hi] = v2;
        }
        __syncthreads();

        // ---- block partial -> global slot (double-buffered, no zeroing needed) ----
        if (tid < 16) {
            const float ssum = wgSum[tid] + wgSum[16+tid] + wgSum[32+tid] + wgSum[48+tid];
            __hip_atomic_store(&partial[(t & 1)*(NBLK*16) + blockIdx.x*16 + tid], ssum,
                               __ATOMIC_RELEASE, __HIP_MEMORY_SCOPE_AGENT);
        }
        // ---- 16-block barrier: monotonic arrive counter + acquire spin ----
        if (tid == 0) {
            __hip_atomic_fetch_add(cnt, 1u, __ATOMIC_ACQ_REL, __HIP_MEMORY_SCOPE_AGENT);
            const unsigned tgt = (unsigned)NBLK * (unsigned)(t + 1);
            while (__hip_atomic_load(cnt, __ATOMIC_ACQUIRE, __HIP_MEMORY_SCOPE_AGENT) < tgt)
                __builtin_amdgcn_s_sleep(1);
        }
        __syncthreads();

        // ---- total sumsq over all C; 1/(sqrt+eps) per q ----
        if (tid < 16) {
            float S = 0.0f;
#pragma unroll
            for (int b = 0; b < NBLK; ++b)
                S += __hip_atomic_load(&partial[(t & 1)*(NBLK*16) + b*16 + tid],
                                       __ATOMIC_RELAXED, __HIP_MEMORY_SCOPE_AGENT);
            nrmb[tid] = 1.0f / (sqrtf(S) + EPS_V);
        }
        __syncthreads();

        // ---- att = att_hat * factor[row]; store canonical to LDS for next step ----
#pragma unroll
        for (int r = 0; r < 8; ++r) {
            const float fac = nrmb[r + 8*hi];        // same-address LDS broadcast
            ATT[(r + 8*hi)*ATT_PITCH + ln] = ah[r] * fac;
        }
        // ATT is wave-private; in-wave DS ordering covers next iteration's reads
    }
}

extern "C" void kernel_launch(void* const* d_in, const int* in_sizes, int n_in,
                              void* d_out, int out_size, void* d_ws, size_t ws_size,
                              hipStream_t stream) {
    (void)in_sizes; (void)n_in; (void)out_size; (void)ws_size;
    const float* K = (const float*)d_in[0];
    const float* Q = (const float*)d_in[1];
    const float* V = (const float*)d_in[2];
    float* out = (float*)d_out;

    // workspace: partial[2][NBLK][16] floats (2048 B) then arrive counter
    float*    partial = (float*)d_ws;
    unsigned* cnt     = (unsigned*)((char*)d_ws + 2048);
    // zero counter (and partial region) each launch: deterministic, capture-safe
    (void)hipMemsetAsync(d_ws, 0, 4096, stream);

    const size_t shmem = (size_t)BLK_LDS * sizeof(float);  // ~126 KB (<320 KB/WGP)
    (void)hipFuncSetAttribute((const void*)rca_kernel,
                              hipFuncAttributeMaxDynamicSharedMemorySize, (int)shmem);
    rca_kernel<<<NBLK, NTHR, shmem, stream>>>(K, Q, V, out, partial, cnt);
}